// DifferentiableRobotModel_25168508354607
// MI455X (gfx1250) — compile-verified
//
#include <hip/hip_runtime.h>
#include <math.h>

#define G_ACC 9.81f

// ---------- small vector helpers (all FMA-shaped) ----------
struct V3 { float x, y, z; };

__device__ __forceinline__ V3 mkv3(float a, float b, float c) { V3 r; r.x=a; r.y=b; r.z=c; return r; }
__device__ __forceinline__ V3 operator+(V3 a, V3 b) { return mkv3(a.x+b.x, a.y+b.y, a.z+b.z); }
__device__ __forceinline__ V3 operator-(V3 a, V3 b) { return mkv3(a.x-b.x, a.y-b.y, a.z-b.z); }
__device__ __forceinline__ V3 operator-(V3 a)       { return mkv3(-a.x, -a.y, -a.z); }
__device__ __forceinline__ V3 operator*(float s, V3 a) { return mkv3(s*a.x, s*a.y, s*a.z); }
__device__ __forceinline__ float dot3(V3 a, V3 b) { return fmaf(a.x,b.x, fmaf(a.y,b.y, a.z*b.z)); }
__device__ __forceinline__ V3 cross3(V3 a, V3 b) {
  return mkv3(fmaf(a.y,b.z,-a.z*b.y), fmaf(a.z,b.x,-a.x*b.z), fmaf(a.x,b.y,-a.y*b.x));
}
// R is 3x3 row-major in 9 floats
__device__ __forceinline__ V3 mv3(const float* R, V3 v) {   // R @ v
  return mkv3(fmaf(R[0],v.x, fmaf(R[1],v.y, R[2]*v.z)),
              fmaf(R[3],v.x, fmaf(R[4],v.y, R[5]*v.z)),
              fmaf(R[6],v.x, fmaf(R[7],v.y, R[8]*v.z)));
}
__device__ __forceinline__ V3 mtv3(const float* R, V3 v) {  // R^T @ v
  return mkv3(fmaf(R[0],v.x, fmaf(R[3],v.y, R[6]*v.z)),
              fmaf(R[1],v.x, fmaf(R[4],v.y, R[7]*v.z)),
              fmaf(R[2],v.x, fmaf(R[5],v.y, R[8]*v.z)));
}

// ---------- LDS constant-table layout (floats) ----------
#define OF_ROT 0    // rot_fix   : 8*9 = 72
#define OF_TRN 72   // trans_fix : 8*3 = 24
#define OF_AX  96   // joint_axes: 7*3 = 21
#define OF_MAS 117  // mass      : 8
#define OF_COM 125  // com       : 8*3 = 24
#define OF_INE 149  // inertia   : 8*9 = 72
#define OF_DMP 221  // damping   : 7
#define CST_N  228

#if defined(__HIP_DEVICE_COMPILE__) && __has_builtin(__builtin_amdgcn_global_load_async_to_lds_b32)
#define HAVE_ASYNC_LDS 1
#else
#define HAVE_ASYNC_LDS 0
#endif

__global__ __launch_bounds__(256) void rnea7_kernel(
    const float* __restrict__ q, const float* __restrict__ qd,
    const float* __restrict__ qdd, const float* __restrict__ rot_fix,
    const float* __restrict__ trans_fix, const float* __restrict__ jaxes,
    const float* __restrict__ mass, const float* __restrict__ com,
    const float* __restrict__ inertia, const float* __restrict__ damping,
    float* __restrict__ out, int batch)
{
  __shared__ float cst[CST_N];

  // ---- stage link constants into LDS via CDNA5 async global->LDS path ----
  {
    const int t = threadIdx.x;
    if (t < CST_N) {
      const float* src;
      if      (t < OF_TRN) src = rot_fix   + t;
      else if (t < OF_AX ) src = trans_fix + (t - OF_TRN);
      else if (t < OF_MAS) src = jaxes     + (t - OF_AX);
      else if (t < OF_COM) src = mass      + (t - OF_MAS);
      else if (t < OF_INE) src = com       + (t - OF_COM);
      else if (t < OF_DMP) src = inertia   + (t - OF_INE);
      else                 src = damping   + (t - OF_DMP);
#if HAVE_ASYNC_LDS
      __builtin_amdgcn_global_load_async_to_lds_b32(
          (__attribute__((address_space(1))) int*)(src),
          (__attribute__((address_space(3))) int*)(&cst[t]),
          0, 0);
#else
      cst[t] = *src;
#endif
    }
#if HAVE_ASYNC_LDS
#if __has_builtin(__builtin_amdgcn_s_wait_asynccnt)
    __builtin_amdgcn_s_wait_asynccnt(0);
#else
    asm volatile("s_wait_asynccnt 0" ::: "memory");
#endif
#endif
    __syncthreads();
  }

  const int i = blockIdx.x * blockDim.x + threadIdx.x;
  if (i >= batch) return;

  // ---- per-element inputs (7 contiguous floats each) ----
  float qv[7], qdv[7], qddv[7];
  {
    const float* qb   = q   + (size_t)i * 7;
    const float* qdb  = qd  + (size_t)i * 7;
    const float* qddb = qdd + (size_t)i * 7;
#pragma unroll
    for (int d = 0; d < 7; ++d) { qv[d] = qb[d]; qdv[d] = qdb[d]; qddv[d] = qddb[d]; }
  }

  float Rm[7][9];      // per-link rotation (kept in VGPRs, reused by backward pass)
  V3 fl[7], fa[7];     // per-link spatial forces

  V3 vl = mkv3(0.f, 0.f, 0.f);
  V3 va = mkv3(0.f, 0.f, 0.f);
  V3 al = mkv3(0.f, 0.f, G_ACC);
  V3 aa = mkv3(0.f, 0.f, 0.f);

  // ---- forward recursion + per-link inertia forces ----
#pragma unroll
  for (int d = 0; d < 7; ++d) {
    const V3 ax = mkv3(cst[OF_AX+3*d], cst[OF_AX+3*d+1], cst[OF_AX+3*d+2]);

    // q ~ N(0,1): well inside the accurate range of the native hardware
    // v_sin_f32/v_cos_f32 path; avoids OCML Payne-Hanek slow paths.
    float s, c;
    __sincosf(qv[d], &s, &c);
    const float omc = 1.0f - c;
    const float n2  = dot3(ax, ax);   // axes are unit, but stay general

    // Rodrigues: Rq = I + s*K + (1-c)*(a a^T - (a.a) I)
    float Rq[9];
    Rq[0] = 1.0f + omc*(ax.x*ax.x - n2);
    Rq[1] = fmaf(-s, ax.z, omc*ax.x*ax.y);
    Rq[2] = fmaf( s, ax.y, omc*ax.x*ax.z);
    Rq[3] = fmaf( s, ax.z, omc*ax.x*ax.y);
    Rq[4] = 1.0f + omc*(ax.y*ax.y - n2);
    Rq[5] = fmaf(-s, ax.x, omc*ax.y*ax.z);
    Rq[6] = fmaf(-s, ax.y, omc*ax.x*ax.z);
    Rq[7] = fmaf( s, ax.x, omc*ax.y*ax.z);
    Rq[8] = 1.0f + omc*(ax.z*ax.z - n2);

    // R = rot_fix[d+1] @ Rq
    const int rb = OF_ROT + 9*(d+1);
    float* R = Rm[d];
#pragma unroll
    for (int r = 0; r < 3; ++r) {
      const float f0 = cst[rb+3*r+0], f1 = cst[rb+3*r+1], f2 = cst[rb+3*r+2];
#pragma unroll
      for (int cc = 0; cc < 3; ++cc)
        R[3*r+cc] = fmaf(f0, Rq[cc], fmaf(f1, Rq[3+cc], f2*Rq[6+cc]));
    }

    const V3 p  = mkv3(cst[OF_TRN+3*(d+1)], cst[OF_TRN+3*(d+1)+1], cst[OF_TRN+3*(d+1)+2]);
    const V3 jv = qdv[d]  * ax;
    const V3 ja = qddv[d] * ax;

    const V3 pinv = -mtv3(R, p);
    const V3 va_n = mtv3(R, va);
    const V3 vl_n = mtv3(R, vl) + cross3(pinv, va_n);
    const V3 va_i = va_n + jv;
    const V3 vl_i = vl_n;
    const V3 aa_n = mtv3(R, aa);
    const V3 al_n = mtv3(R, al) + cross3(pinv, aa_n);
    const V3 aa_i = aa_n + ja + cross3(va_i, jv);
    const V3 al_i = al_n + cross3(vl_i, jv);

    vl = vl_i; va = va_i; al = al_i; aa = aa_i;

    // spatial inertia: Io = I + m*((c.c)I - c c^T)
    const float m  = cst[OF_MAS + d + 1];
    const V3 cm    = mkv3(cst[OF_COM+3*(d+1)], cst[OF_COM+3*(d+1)+1], cst[OF_COM+3*(d+1)+2]);
    const V3 mc    = m * cm;
    const float c2 = dot3(cm, cm);
    const int ib   = OF_INE + 9*(d+1);
    float Io[9];
    Io[0] = cst[ib+0] + m*(c2 - cm.x*cm.x);
    Io[1] = cst[ib+1] - m*cm.x*cm.y;
    Io[2] = cst[ib+2] - m*cm.x*cm.z;
    Io[3] = cst[ib+3] - m*cm.y*cm.x;
    Io[4] = cst[ib+4] + m*(c2 - cm.y*cm.y);
    Io[5] = cst[ib+5] - m*cm.y*cm.z;
    Io[6] = cst[ib+6] - m*cm.z*cm.x;
    Io[7] = cst[ib+7] - m*cm.z*cm.y;
    Io[8] = cst[ib+8] + m*(c2 - cm.z*cm.z);

    const V3 Ia_l = m*al_i + cross3(aa_i, mc);
    const V3 Ia_a = mv3(Io, aa_i) + cross3(mc, al_i);
    const V3 Iv_l = m*vl_i + cross3(va_i, mc);
    const V3 Iv_a = mv3(Io, va_i) + cross3(mc, vl_i);

    fl[d] = Ia_l + cross3(va_i, Iv_l);
    fa[d] = Ia_a + cross3(va_i, Iv_a) + cross3(vl_i, Iv_l);
  }

  // ---- backward recursion ----
  V3 cl = mkv3(0.f, 0.f, 0.f);
  V3 ca = mkv3(0.f, 0.f, 0.f);
  float* ob = out + (size_t)i * 7;
#pragma unroll
  for (int d = 6; d >= 0; --d) {
    const V3 tl = fl[d] + cl;
    const V3 ta = fa[d] + ca;
    const V3 ax = mkv3(cst[OF_AX+3*d], cst[OF_AX+3*d+1], cst[OF_AX+3*d+2]);
    const float tau = dot3(ta, ax);
    const float* R = Rm[d];
    const V3 p  = mkv3(cst[OF_TRN+3*(d+1)], cst[OF_TRN+3*(d+1)+1], cst[OF_TRN+3*(d+1)+2]);
    const V3 nl = mv3(R, tl);
    const V3 na = mv3(R, ta) + cross3(p, nl);
    cl = nl; ca = na;
    ob[d] = fmaf(cst[OF_DMP+d], qdv[d], tau);
  }
}

extern "C" void kernel_launch(void* const* d_in, const int* in_sizes, int n_in,
                              void* d_out, int out_size, void* d_ws, size_t ws_size,
                              hipStream_t stream) {
  const float* q         = (const float*)d_in[0];
  const float* qd        = (const float*)d_in[1];
  const float* qdd_des   = (const float*)d_in[2];
  const float* rot_fix   = (const float*)d_in[3];
  const float* trans_fix = (const float*)d_in[4];
  const float* jaxes     = (const float*)d_in[5];
  const float* mass      = (const float*)d_in[6];
  const float* com       = (const float*)d_in[7];
  const float* inertia   = (const float*)d_in[8];
  const float* damping   = (const float*)d_in[9];
  float* out = (float*)d_out;

  const int batch   = in_sizes[0] / 7;
  const int threads = 256;                       // 8 wave32s per workgroup
  const int blocks  = (batch + threads - 1) / threads;

  hipLaunchKernelGGL(rnea7_kernel, dim3(blocks), dim3(threads), 0, stream,
                     q, qd, qdd_des, rot_fix, trans_fix, jaxes, mass, com,
                     inertia, damping, out, batch);
}